// SamplingLoss_31104153158230
// MI455X (gfx1250) — compile-verified
//
#include <hip/hip_runtime.h>
#include <math.h>

typedef __attribute__((ext_vector_type(2))) float v2f;
typedef __attribute__((ext_vector_type(8))) float v8f;

#define IMG_H 1024
#define IMG_W 2048
#define PI_F 3.14159265358979323846f

__global__ __launch_bounds__(256) void sampling_loss_main(
    const float* __restrict__ trans,
    const float* __restrict__ yawp,
    const float* __restrict__ pitchp,
    const float* __restrict__ rollp,
    const float* __restrict__ xyz,
    const float* __restrict__ rgb,
    const float* __restrict__ img,
    const float* __restrict__ imgw,
    const float* __restrict__ pcdw,
    float* __restrict__ partials,
    int nPts)
{
    const int lane = threadIdx.x & 31;
    const int wib  = threadIdx.x >> 5;
    const int wavesPerBlock = blockDim.x >> 5;
    const long long waveGlobal = (long long)blockIdx.x * wavesPerBlock + wib;
    const long long waveStride = (long long)gridDim.x * wavesPerBlock;

    // ---- rotation matrix R = RZ(yaw) @ RY(pitch) @ RX(roll), b = -R @ t ----
    float cy = cosf(yawp[0]),  sy = sinf(yawp[0]);
    float cp = cosf(pitchp[0]), sp = sinf(pitchp[0]);
    float cr = cosf(rollp[0]),  sr = sinf(rollp[0]);
    float r00 = cy*cp, r01 = -sy*cr + cy*sp*sr, r02 =  sy*sr + cy*sp*cr;
    float r10 = sy*cp, r11 =  cy*cr + sy*sp*sr, r12 = -cy*sr + sy*sp*cr;
    float r20 = -sp,   r21 =  cp*sr,            r22 =  cp*cr;
    float t0 = trans[0], t1 = trans[1], t2 = trans[2];
    float b0 = -(r00*t0 + r01*t1 + r02*t2);
    float b1 = -(r10*t0 + r11*t1 + r12*t2);
    float b2 = -(r20*t0 + r21*t1 + r22*t2);

    // ---- A fragment for V_WMMA_F32_16X16X4_F32: A = [R | -R t], rows 3..15 zero
    // 32-bit A 16x4 layout: lanes 0-15 = A[M][K=0..1]; lanes 16-31 = A[M][K=2..3]
    const int  hl = lane & 15;
    const bool hi = lane >= 16;
    float a0, a1;
    if (!hi) {
        a0 = (hl==0) ? r00 : (hl==1) ? r10 : (hl==2) ? r20 : 0.0f;
        a1 = (hl==0) ? r01 : (hl==1) ? r11 : (hl==2) ? r21 : 0.0f;
    } else {
        a0 = (hl==0) ? r02 : (hl==1) ? r12 : (hl==2) ? r22 : 0.0f;
        a1 = (hl==0) ? b0  : (hl==1) ? b1  : (hl==2) ? b2  : 0.0f;
    }
    v2f afrag; afrag[0] = a0; afrag[1] = a1;

    float lsum = 0.0f, csum = 0.0f;
    const int pmax = nPts - 1;

    for (long long base = waveGlobal * 32; base < nPts; base += waveStride * 32) {
        // B = [x; y; z; 1], 16 point-columns per WMMA; two batches cover 32 points.
        // 32-bit B 4x16 layout: lanes 0-15 = rows K=0,1 ; lanes 16-31 = rows K=2,3
        long long p1 = base + hl;        if (p1 > pmax) p1 = pmax;
        long long p2 = base + 16 + hl;   if (p2 > pmax) p2 = pmax;
        v2f bf1, bf2;
        if (!hi) {
            bf1[0] = xyz[p1*3+0]; bf1[1] = xyz[p1*3+1];
            bf2[0] = xyz[p2*3+0]; bf2[1] = xyz[p2*3+1];
        } else {
            bf1[0] = xyz[p1*3+2]; bf1[1] = 1.0f;
            bf2[0] = xyz[p2*3+2]; bf2[1] = 1.0f;
        }
        v8f czero = {};
        v8f d1 = __builtin_amdgcn_wmma_f32_16x16x4_f32(false, afrag, false, bf1,
                                                       (short)0, czero, false, false);
        v8f d2 = __builtin_amdgcn_wmma_f32_16x16x4_f32(false, afrag, false, bf2,
                                                       (short)0, czero, false, false);

        // D rows 0/1/2 (lanes 0-15) = rotated x/y/z. Move batch-2 results into
        // lanes 16-31 so every lane owns one point for the gather phase.
        float hx = __shfl(d2[0], hl, 32);
        float hy = __shfl(d2[1], hl, 32);
        float hz = __shfl(d2[2], hl, 32);
        float px = hi ? hx : d1[0];
        float py = hi ? hy : d1[1];
        float pz = hi ? hz : d1[2];

        // ---- spherical projection ----
        float phi   = atan2f(py, px) + PI_F;
        float theta = atan2f(sqrtf(px*px + py*py), pz);
        float cxn = 1.0f - phi * (1.0f/PI_F);          // 2*(1 - phi/2pi) - 1
        float cyn = theta * (2.0f/PI_F) - 1.0f;        // 2*theta/pi - 1

        // ---- bilinear sample (align_corners=False, border clamp) ----
        float xx = fminf(fmaxf((cxn + 1.0f)*0.5f*(float)IMG_W - 0.5f, 0.0f), (float)(IMG_W-1));
        float yy = fminf(fmaxf((cyn + 1.0f)*0.5f*(float)IMG_H - 0.5f, 0.0f), (float)(IMG_H-1));
        float fx0 = floorf(xx), fy0 = floorf(yy);
        float wx = xx - fx0, wy = yy - fy0;
        int x0 = (int)fx0, y0 = (int)fy0;
        int x1 = min(x0+1, IMG_W-1), y1 = min(y0+1, IMG_H-1);
        float w00 = (1.0f-wx)*(1.0f-wy);
        float w01 = wx*(1.0f-wy);
        float w10 = (1.0f-wx)*wy;
        float w11 = wx*wy;
        int i00 = y0*IMG_W + x0, i01 = y0*IMG_W + x1;
        int i10 = y1*IMG_W + x0, i11 = y1*IMG_W + x1;

        float s0 = img[i00*3+0]*w00 + img[i01*3+0]*w01 + img[i10*3+0]*w10 + img[i11*3+0]*w11;
        float s1 = img[i00*3+1]*w00 + img[i01*3+1]*w01 + img[i10*3+1]*w10 + img[i11*3+1]*w11;
        float s2 = img[i00*3+2]*w00 + img[i01*3+2]*w01 + img[i10*3+2]*w10 + img[i11*3+2]*w11;
        float wimg = imgw[i00]*w00 + imgw[i01]*w01 + imgw[i10]*w10 + imgw[i11]*w11;

        // ---- loss ----
        long long p = base + lane;
        long long pc = p > pmax ? pmax : p;
        float g0 = rgb[pc*3+0], g1 = rgb[pc*3+1], g2 = rgb[pc*3+2];
        float pw = pcdw[pc];
        float e0 = s0-g0, e1 = s1-g1, e2 = s2-g2;
        float raw = sqrtf(e0*e0 + e1*e1 + e2*e2);
        bool m = !((s0==0.0f) && (s1==0.0f) && (s2==0.0f)) && (p < nPts);
        float loss = 0.5f*(wimg + pw)*raw;
        lsum += m ? loss : 0.0f;
        csum += m ? 1.0f : 0.0f;
    }

    // ---- deterministic reduction: wave shfl tree -> LDS -> per-block partial ----
    for (int off = 16; off > 0; off >>= 1) {
        lsum += __shfl_down(lsum, off, 32);
        csum += __shfl_down(csum, off, 32);
    }
    __shared__ float sL[8], sC[8];
    if (lane == 0) { sL[wib] = lsum; sC[wib] = csum; }
    __syncthreads();
    if (threadIdx.x == 0) {
        float L = 0.0f, C = 0.0f;
        for (int i = 0; i < wavesPerBlock; ++i) { L += sL[i]; C += sC[i]; }
        partials[2*blockIdx.x]   = L;
        partials[2*blockIdx.x+1] = C;
    }
}

__global__ __launch_bounds__(256) void sampling_loss_finalize(
    const float* __restrict__ partials, int nBlocks, float* __restrict__ out)
{
    float L = 0.0f, C = 0.0f;
    for (int i = threadIdx.x; i < nBlocks; i += blockDim.x) {
        L += partials[2*i];
        C += partials[2*i+1];
    }
    for (int off = 16; off > 0; off >>= 1) {
        L += __shfl_down(L, off, 32);
        C += __shfl_down(C, off, 32);
    }
    __shared__ float sL[8], sC[8];
    int lane = threadIdx.x & 31, wib = threadIdx.x >> 5;
    if (lane == 0) { sL[wib] = L; sC[wib] = C; }
    __syncthreads();
    if (threadIdx.x == 0) {
        float tL = 0.0f, tC = 0.0f;
        int nw = blockDim.x >> 5;
        for (int i = 0; i < nw; ++i) { tL += sL[i]; tC += sC[i]; }
        out[0] = tL / tC;
    }
}

extern "C" void kernel_launch(void* const* d_in, const int* in_sizes, int n_in,
                              void* d_out, int out_size, void* d_ws, size_t ws_size,
                              hipStream_t stream) {
    const float* trans = (const float*)d_in[0];
    const float* yaw   = (const float*)d_in[1];
    const float* pitch = (const float*)d_in[2];
    const float* roll  = (const float*)d_in[3];
    const float* xyz   = (const float*)d_in[4];
    const float* rgb   = (const float*)d_in[5];
    const float* img   = (const float*)d_in[6];
    const float* imgw  = (const float*)d_in[7];
    const float* pcdw  = (const float*)d_in[8];
    float* out      = (float*)d_out;
    float* partials = (float*)d_ws;
    int nPts = in_sizes[4] / 3;

    int G = 1024;  // 8192 waves: ~7.6 chunks of 32 pts per wave; partials = 8 KB
    size_t need = (size_t)G * 2 * sizeof(float);
    if (ws_size < need) {
        G = (int)(ws_size / (2 * sizeof(float)));
        if (G < 1) G = 1;
    }

    sampling_loss_main<<<G, 256, 0, stream>>>(trans, yaw, pitch, roll, xyz, rgb,
                                              img, imgw, pcdw, partials, nPts);
    sampling_loss_finalize<<<1, 256, 0, stream>>>(partials, G, out);
}